// HP_22282290332071
// MI455X (gfx1250) — compile-verified
//
#include <hip/hip_runtime.h>
#include <hip/hip_bf16.h>

typedef __attribute__((ext_vector_type(2))) float v2f;
typedef __attribute__((ext_vector_type(8))) float v8f;

#define NTYPES 23      // type values 0..22 (0 == pad)
#define ACOLS  24      // alpha is (23, 24)
#define L1 1025
#define L  1024
#define NB 32
#define RTILES 64      // 1024 rows / 16

// workspace float layout
#define WS_OMEGA   0
#define WS_SUMMU1  1
#define WS_AHAT    2              // 552 floats
#define WS_MHAT    (2 + 552)      // 23 floats
#define WS_COLSUM  (2 + 552 + 23) // 24 floats
#define WS_PART    640            // 64*32 = 2048 partials (-sum log rate per tile)
#define WS_COMP    2700           // 32 per-batch compensators

#define LOG2E 1.4426950408889634f

// raw v_exp_f32 (no denormal-range fixup; underflow flushes to 0, which is
// exactly what we want for exp(-w*dt) contributions below 1e-38)
__device__ __forceinline__ float fast_exp2(float x) {
    return __builtin_amdgcn_exp2f(x);
}

__device__ __forceinline__ float softplusf(float x) {
    return (x > 20.0f) ? x : log1pf(__expf(x));
}

// ---------------------------------------------------------------------------
// Kernel 0: softplus of mu/alpha/beta + reductions (tiny; single thread)
// ---------------------------------------------------------------------------
__global__ void hk_precompute(const float* alpha, const float* beta,
                              const float* mu, float* ws) {
    ws[WS_OMEGA] = softplusf(beta[0]);
    float sm = 0.0f;
    for (int k = 0; k < NTYPES; ++k) {
        float m = softplusf(mu[k]);
        ws[WS_MHAT + k] = m;
        if (k >= 1) sm += m;
    }
    ws[WS_SUMMU1] = sm;
    for (int i = 0; i < NTYPES * ACOLS; ++i)
        ws[WS_AHAT + i] = softplusf(alpha[i]);
    for (int c = 0; c < ACOLS; ++c) {
        float s = 0.0f;
        for (int k = 1; k < NTYPES; ++k) s += ws[WS_AHAT + k * ACOLS + c];
        ws[WS_COLSUM + c] = s;
    }
}

// ---------------------------------------------------------------------------
// Kernel 1: triangular rates via WMMA row-reduction.
// grid = (64 row tiles, 32 batches), block = 32 (one wave).
//
// Inner loop is fully branchless:
//  - sAz has row 0 and column 0 zeroed -> pad masking is free via the gather.
//  - sPK interleaves {K*T_j , tj*4} so one ds_load_b128 feeds 2 columns.
//  - exp(-w*(Ti-Tj)) == exp2(K*Tj - K*Ti), K = w*log2(e): v_sub + bare v_exp.
// ---------------------------------------------------------------------------
__global__ __launch_bounds__(32)
void hk_rates(const int* __restrict__ type_seq,
              const float* __restrict__ time_seq,
              const float* __restrict__ ws,
              float* __restrict__ partials) {
    const int r    = blockIdx.x;      // row tile
    const int b    = blockIdx.y;      // batch
    const int lane = threadIdx.x;     // 0..31

    __shared__ float sPK[2 * (L1 + 1)];        // [2j]=K*T_j, [2j+1]=bits(tj*4)
    __shared__ float sAz[NTYPES * ACOLS];      // Ahat with row0/col0 zeroed
    __shared__ int   rowType[16];
    __shared__ float red[2];

    const int*   tsq = type_seq + b * L1;
    const float* tms = time_seq + b * L1;
    const float omega = ws[WS_OMEGA];
    const float K = omega * LOG2E;

    for (int k = lane; k < L1; k += 32) {
        sPK[2 * k] = K * tms[k];
        ((int*)sPK)[2 * k + 1] = tsq[k] << 2;  // byte offset into Ahat row
    }
    for (int k = lane; k < NTYPES * ACOLS; k += 32) {
        const int rowk = k / ACOLS;
        const int colk = k - rowk * ACOLS;
        sAz[k] = (rowk == 0 || colk == 0) ? 0.0f : ws[WS_AHAT + k];
    }
    __syncthreads();

    const int row  = lane & 15;       // A-fragment row M = lane%16
    const int half = lane >> 4;       // K-pair select: 0 -> K{0,1}, 1 -> K{2,3}
    const int ig   = 1 + 16 * r + row;              // event index 1..1024
    const int tiB  = ((const int*)sPK)[2 * ig + 1]; // ti*4
    const float eRow = sPK[2 * ig];                 // K*T_i
    if (half == 0) rowType[row] = tiB >> 2;
    const char* ArowB = (const char*)sAz + tiB * ACOLS; // &sAz[ti*24] in bytes

    // B (4x16): one-hot column 0 -> D[:,0] = row sums.
    v2f bm;
    {
        float bv = (row == 0) ? 1.0f : 0.0f;
        bm[0] = bv; bm[1] = bv;
    }

    v8f acc = {};

    // ---- full tiles: every j < ig guaranteed ----
    for (int c = 0; c < r; ++c) {
        const int base = 16 * c + 2 * half;
#pragma unroll
        for (int q = 0; q < 4; ++q) {
            const int j0 = base + 4 * q;
            const float4 pk = *(const float4*)(sPK + 2 * j0); // b128, 16B aligned
            const int tj0 = __float_as_int(pk.y);
            const int tj1 = __float_as_int(pk.w);
            const float a0 = *(const float*)(ArowB + tj0);
            const float a1 = *(const float*)(ArowB + tj1);
            v2f a;
            a[0] = a0 * fast_exp2(pk.x - eRow);
            a[1] = a1 * fast_exp2(pk.z - eRow);
            acc = __builtin_amdgcn_wmma_f32_16x16x4_f32(
                false, a, false, bm, (short)0, acc, false, false);
        }
    }

    // ---- diagonal tile c == r: mask j < ig via float select (no branch) ----
    {
        const int base = 16 * r + 2 * half;
#pragma unroll
        for (int q = 0; q < 4; ++q) {
            const int j0 = base + 4 * q;
            const float4 pk = *(const float4*)(sPK + 2 * j0);
            const int tj0 = __float_as_int(pk.y);
            const int tj1 = __float_as_int(pk.w);
            const float m0 = (j0 < ig)     ? 1.0f : 0.0f;
            const float m1 = (j0 + 1 < ig) ? 1.0f : 0.0f;
            const float a0 = *(const float*)(ArowB + tj0);
            const float a1 = *(const float*)(ArowB + tj1);
            v2f a;
            a[0] = m0 * a0 * fast_exp2(pk.x - eRow);
            a[1] = m1 * a1 * fast_exp2(pk.z - eRow);
            acc = __builtin_amdgcn_wmma_f32_16x16x4_f32(
                false, a, false, bm, (short)0, acc, false, false);
        }
    }
    __syncthreads();

    // D layout: VGPR v, lanes 0-15 -> M=v, lanes 16-31 -> M=v+8; N = lane%16.
    // Column 0 (the row sums) lives in lane 0 (rows 0..7) and lane 16 (8..15).
    float part = 0.0f;
    if ((lane & 15) == 0) {
#pragma unroll
        for (int v = 0; v < 8; ++v) {
            const int rr = v + 8 * half;
            const int t  = rowType[rr];
            if (t != 0) {
                float rate = ws[WS_MHAT + t] + omega * acc[v];
                rate = fmaxf(rate, 1e-10f);
                part -= __logf(rate);
            }
        }
    }
    if (lane == 0)  red[0] = part;
    if (lane == 16) red[1] = part;
    __syncthreads();
    if (lane == 0) partials[b * RTILES + r] = red[0] + red[1];
}

// ---------------------------------------------------------------------------
// Kernel 2: per-batch compensator. grid = 32 blocks of 256.
// ---------------------------------------------------------------------------
__global__ void hk_comp(const int* __restrict__ type_seq,
                        const float* __restrict__ time_seq,
                        const float* __restrict__ ws,
                        float* __restrict__ compOut) {
    const int b   = blockIdx.x;
    const int tid = threadIdx.x;
    const float omega = ws[WS_OMEGA];
    // time_seq is a cumsum of non-negative gaps -> max is last element
    const float maxT = time_seq[b * L1 + (L1 - 1)];
    const float K = omega * LOG2E;
    float local = 0.0f;
    for (int j = tid; j < L1; j += blockDim.x) {
        const int t = type_seq[b * L1 + j];
        const float ik = 1.0f - fast_exp2(K * (time_seq[b * L1 + j] - maxT));
        local += (t != 0) ? ws[WS_COLSUM + t] * ik : 0.0f;
    }
    __shared__ float red[256];
    red[tid] = local;
    __syncthreads();
    for (int s = 128; s > 0; s >>= 1) {
        if (tid < s) red[tid] += red[tid + s];
        __syncthreads();
    }
    if (tid == 0) compOut[b] = red[0] + maxT * ws[WS_SUMMU1];
}

// ---------------------------------------------------------------------------
// Kernel 3: deterministic final reduction into d_out[0].
// ---------------------------------------------------------------------------
__global__ void hk_final(const float* __restrict__ partials,
                         const float* __restrict__ comps,
                         float* __restrict__ out) {
    const int tid = threadIdx.x;
    float local = 0.0f;
    for (int k = tid; k < NB * RTILES; k += 256) local += partials[k];
    for (int k = tid; k < NB; k += 256)          local += comps[k];
    __shared__ float red[256];
    red[tid] = local;
    __syncthreads();
    for (int s = 128; s > 0; s >>= 1) {
        if (tid < s) red[tid] += red[tid + s];
        __syncthreads();
    }
    if (tid == 0) out[0] = red[0];
}

extern "C" void kernel_launch(void* const* d_in, const int* in_sizes, int n_in,
                              void* d_out, int out_size, void* d_ws, size_t ws_size,
                              hipStream_t stream) {
    const int*   type_seq = (const int*)d_in[0];
    const float* time_seq = (const float*)d_in[1];
    const float* alpha    = (const float*)d_in[2];
    const float* beta     = (const float*)d_in[3];
    const float* mu       = (const float*)d_in[4];
    float* out = (float*)d_out;
    float* ws  = (float*)d_ws;

    hk_precompute<<<1, 1, 0, stream>>>(alpha, beta, mu, ws);
    hk_rates<<<dim3(RTILES, NB), 32, 0, stream>>>(type_seq, time_seq, ws,
                                                  ws + WS_PART);
    hk_comp<<<NB, 256, 0, stream>>>(type_seq, time_seq, ws, ws + WS_COMP);
    hk_final<<<1, 256, 0, stream>>>(ws + WS_PART, ws + WS_COMP, out);
}